// PostProcessor_33784212750804
// MI455X (gfx1250) — compile-verified
//
#include <hip/hip_runtime.h>
#include <hip/hip_bf16.h>
#include <math.h>

typedef float v2f __attribute__((ext_vector_type(2)));
typedef float v8f __attribute__((ext_vector_type(8)));

#define BB     4
#define PRE    512
#define POST   100
#define NPROB  16      // 4 variants x 4 batches
#define CAND   2048
#define IOU_TH 0.01f

// ---------- helpers ----------
__device__ __forceinline__ unsigned fkey(float f) {
  unsigned u = __float_as_uint(f);
  return (u & 0x80000000u) ? ~u : (u | 0x80000000u);  // monotonic float->uint
}

// ---------- zero scratch ----------
__global__ void k_zero(unsigned* p, int n) {
  for (int i = blockIdx.x * blockDim.x + threadIdx.x; i < n; i += gridDim.x * blockDim.x)
    p[i] = 0u;
}

// ---------- scores: sigmoid(sigmoid(x)) ----------
__global__ void k_scores(const float* __restrict__ logits, float* __restrict__ tot, int BN) {
  for (int i = blockIdx.x * blockDim.x + threadIdx.x; i < BN; i += gridDim.x * blockDim.x) {
    float s = 1.f / (1.f + __expf(-logits[i]));
    tot[i]  = 1.f / (1.f + __expf(-s));
  }
}

// ---------- top-512 select: level-1 histogram (key>>24) ----------
__global__ void k_hist1(const float* __restrict__ tot, const float* __restrict__ res,
                        int N, unsigned* __restrict__ hist1) {
  int row = blockIdx.y;            // 0..7 : (scoreset<<2)|batch
  int ss = row >> 2, b = row & 3;
  __shared__ unsigned lh[256];
  for (int i = threadIdx.x; i < 256; i += blockDim.x) lh[i] = 0u;
  __syncthreads();
  const float* src = ss ? (res + (size_t)b * N) : (tot + (size_t)b * N);
  for (int i = blockIdx.x * blockDim.x + threadIdx.x; i < N; i += gridDim.x * blockDim.x)
    atomicAdd(&lh[fkey(src[i]) >> 24], 1u);
  __syncthreads();
  for (int i = threadIdx.x; i < 256; i += blockDim.x)
    if (lh[i]) atomicAdd(&hist1[row * 256 + i], lh[i]);
}

__global__ void k_thresh1(const unsigned* __restrict__ hist1, int* t1, int* cA) {
  int row = blockIdx.x;
  if (threadIdx.x) return;
  unsigned cum = 0; int tb = 0; unsigned above = 0; bool found = false;
  for (int bin = 255; bin >= 0 && !found; --bin) {
    unsigned h = hist1[row * 256 + bin];
    if (cum + h >= PRE) { tb = bin; above = cum; found = true; }
    else cum += h;
  }
  t1[row] = tb;
  cA[row] = (int)above;
}

// ---------- level-2 histogram ((key>>12)&0xFFF) inside threshold bin ----------
__global__ void k_hist2(const float* __restrict__ tot, const float* __restrict__ res,
                        int N, const int* __restrict__ t1, unsigned* __restrict__ hist2) {
  int row = blockIdx.y;
  int ss = row >> 2, b = row & 3;
  __shared__ unsigned lh[4096];
  for (int i = threadIdx.x; i < 4096; i += blockDim.x) lh[i] = 0u;
  __syncthreads();
  const float* src = ss ? (res + (size_t)b * N) : (tot + (size_t)b * N);
  int T1 = t1[row];
  for (int i = blockIdx.x * blockDim.x + threadIdx.x; i < N; i += gridDim.x * blockDim.x) {
    unsigned key = fkey(src[i]);
    if ((int)(key >> 24) == T1) atomicAdd(&lh[(key >> 12) & 0xFFFu], 1u);
  }
  __syncthreads();
  for (int i = threadIdx.x; i < 4096; i += blockDim.x)
    if (lh[i]) atomicAdd(&hist2[row * 4096 + i], lh[i]);
}

__global__ void k_thresh2(const unsigned* __restrict__ hist2, const int* __restrict__ cA, int* t2) {
  int row = blockIdx.x;
  if (threadIdx.x) return;
  int need = PRE - cA[row];
  unsigned cum = 0; int tb = 0; bool found = false;
  for (int bin = 4095; bin >= 0 && !found; --bin) {
    unsigned h = hist2[row * 4096 + bin];
    if ((int)(cum + h) >= need) { tb = bin; found = true; }
    else cum += h;
  }
  t2[row] = found ? tb : 0;
}

// ---------- compact candidates above threshold ----------
__global__ void k_compact(const float* __restrict__ tot, const float* __restrict__ res,
                          int N, const int* __restrict__ t1, const int* __restrict__ t2,
                          int* __restrict__ cnt, unsigned* __restrict__ ckey, int* __restrict__ cidx) {
  int row = blockIdx.y;
  int ss = row >> 2, b = row & 3;
  const float* src = ss ? (res + (size_t)b * N) : (tot + (size_t)b * N);
  int T1 = t1[row], T2 = t2[row];
  for (int i = blockIdx.x * blockDim.x + threadIdx.x; i < N; i += gridDim.x * blockDim.x) {
    unsigned key = fkey(src[i]);
    int hb = (int)(key >> 24);
    bool acc = (hb > T1) || (hb == T1 && (int)((key >> 12) & 0xFFFu) >= T2);
    if (acc) {
      int pos = atomicAdd(&cnt[row], 1);
      if (pos < CAND) { ckey[row * CAND + pos] = key; cidx[row * CAND + pos] = i; }
    }
  }
}

// ---------- bitonic sort candidates (desc key, asc idx), keep top-512 ----------
__global__ void k_sort(const unsigned* __restrict__ ckey, const int* __restrict__ cidx,
                       const int* __restrict__ cnt, int* __restrict__ topI) {
  int row = blockIdx.x;
  __shared__ unsigned long long comp[CAND];
  int c = cnt[row]; if (c > CAND) c = CAND;
  for (int e = threadIdx.x; e < CAND; e += blockDim.x) {
    if (e < c)
      comp[e] = ((unsigned long long)ckey[row * CAND + e] << 32) |
                (unsigned)(~(unsigned)cidx[row * CAND + e]);
    else
      comp[e] = 0ull;   // sinks to bottom (real keys have MSB set)
  }
  __syncthreads();
  for (int k = 2; k <= CAND; k <<= 1) {
    for (int j = k >> 1; j > 0; j >>= 1) {
      for (int e = threadIdx.x; e < CAND; e += blockDim.x) {
        int l = e ^ j;
        if (l > e) {
          bool wantDesc = ((e & k) == 0);
          unsigned long long a = comp[e], bb = comp[l];
          if ((a < bb) == wantDesc) { comp[e] = bb; comp[l] = a; }
        }
      }
      __syncthreads();
    }
  }
  for (int e = threadIdx.x; e < PRE; e += blockDim.x)
    topI[row * PRE + e] = (int)(~(unsigned)(comp[e] & 0xFFFFFFFFull));
}

// ---------- gather + decode selected boxes, build NMS geometry ----------
__global__ void k_decode(const int* __restrict__ topI, const float* __restrict__ breg,
                         const float* __restrict__ anch, int N,
                         float* __restrict__ box7, float* __restrict__ geom) {
  int t = blockIdx.x * blockDim.x + threadIdx.x;
  if (t >= NPROB * PRE) return;
  int p = t >> 9, k = t & (PRE - 1);
  int v = p >> 2, b = p & 3;
  int row = ((v == 3) ? 1 : 0) * 4 + b;
  int idx = topI[row * PRE + k];
  size_t g = (size_t)b * N + (unsigned)idx;
  const float* d = breg + g * 7;
  const float* a = anch + g * 7;
  float dx = d[0] * 0.1f, dy = d[1] * 0.1f, dz = d[2] * 0.1f;
  float dw = d[3] * 0.2f, dl = d[4] * 0.2f, dh = d[5] * 0.2f, dr = d[6] * 0.1f;
  float xa = a[0], ya = a[1], za = a[2], wa = a[3], la = a[4], ha = a[5], ra = a[6];
  float x, y;
  if (v < 2) { float diag = sqrtf(wa * wa + la * la); x = dx * diag + xa; y = dy * diag + ya; }
  else       { x = dx * wa + xa; y = dy * la + ya; }
  float z = dz * ha + za;
  float w = __expf(dw) * wa, l = __expf(dl) * la, h = __expf(dh) * ha;
  float r = dr + ra;
  if (v == 1) r = atan2f(__sinf(r), __cosf(r));
  float* bx = box7 + (size_t)(p * PRE + k) * 7;
  bx[0] = x; bx[1] = y; bx[2] = z; bx[3] = w; bx[4] = l; bx[5] = h; bx[6] = r;
  float cs = __cosf(r), sn = __sinf(r);
  float hw = 0.5f * w, hl = 0.5f * l;
  const float sx[4] = { hw, hw, -hw, -hw };
  const float sy[4] = { -hl, hl, hl, -hl };
  size_t gb = (size_t)p * 13 * PRE;
  geom[gb + 0 * PRE + k] = x;
  geom[gb + 1 * PRE + k] = y;
  geom[gb + 2 * PRE + k] = x * x + y * y;
  geom[gb + 3 * PRE + k] = 0.5f * sqrtf(w * w + l * l);  // circumradius
  geom[gb + 4 * PRE + k] = fabsf(w * l);                 // area
  for (int c = 0; c < 4; ++c) {
    geom[gb + (5 + c) * PRE + k] = x + cs * sx[c] - sn * sy[c];
    geom[gb + (9 + c) * PRE + k] = y + sn * sx[c] + cs * sy[c];
  }
}

// ---------- Sutherland–Hodgman clip area (4-gon vs 4-gon, max 8 verts) ----------
__device__ float interArea(const float* cix, const float* ciy,
                           const float* cjx, const float* cjy) {
  float px[8], py[8];
  int n = 4;
  for (int k = 0; k < 4; ++k) { px[k] = cix[k]; py[k] = ciy[k]; }
  for (int e = 0; e < 4; ++e) {
    float p1x = cjx[e], p1y = cjy[e];
    float p2x = cjx[(e + 1) & 3], p2y = cjy[(e + 1) & 3];
    float ex = p2x - p1x, ey = p2y - p1y;
    float qx[8], qy[8]; int m = 0;
    for (int k = 0; k < n; ++k) {
      int kn = (k + 1 == n) ? 0 : k + 1;
      float dc = ex * (py[k] - p1y) - ey * (px[k] - p1x);
      float dn = ex * (py[kn] - p1y) - ey * (px[kn] - p1x);
      bool ic = dc >= 0.f, in2 = dn >= 0.f;
      if (ic && m < 8) { qx[m] = px[k]; qy[m] = py[k]; ++m; }
      if ((ic != in2) && m < 8) {
        float den = dc - dn;
        den = (fabsf(den) < 1e-8f) ? 1e-8f : den;
        float tt = dc / den;
        qx[m] = px[k] + tt * (px[kn] - px[k]);
        qy[m] = py[k] + tt * (py[kn] - py[k]);
        ++m;
      }
    }
    n = m;
    for (int k = 0; k < m; ++k) { px[k] = qx[k]; py[k] = qy[k]; }
    if (n == 0) return 0.f;
  }
  float cr = 0.f;
  for (int k = 0; k < n; ++k) {
    int kn = (k + 1 == n) ? 0 : k + 1;
    cr += px[k] * py[kn] - py[k] * px[kn];
  }
  return 0.5f * fabsf(cr);
}

// ---------- IOU tiles: WMMA f32 16x16x4 distance prefilter + clip ----------
// dist^2[i][j] = A[i]·B[j] with A=[x,y,x^2+y^2,1], B=[-2x,-2y,1,x^2+y^2] -> K=4 exact fit.
__global__ void k_iou(const float* __restrict__ geom, unsigned* __restrict__ mask) {
  int tile = blockIdx.x;
  int p  = tile >> 10;
  int rs = tile & 1023;
  int i0 = (rs >> 5) * 16;
  int j0 = (rs & 31) * 16;
  if (j0 + 15 <= i0) return;   // whole tile below diagonal (uniform exit, wave intact)

  __shared__ float six[16], siy[16], siq[16], sir[16], sia[16];
  __shared__ float sjx[16], sjy[16], sjq[16], sjr[16], sja[16];
  __shared__ float sicx[16][4], sicy[16][4], sjcx[16][4], sjcy[16][4];

  int lane = threadIdx.x;
  size_t gb = (size_t)p * 13 * PRE;
  if (lane < 16) {
    int s = i0 + lane;
    six[lane] = geom[gb + 0 * PRE + s];
    siy[lane] = geom[gb + 1 * PRE + s];
    siq[lane] = geom[gb + 2 * PRE + s];
    sir[lane] = geom[gb + 3 * PRE + s];
    sia[lane] = geom[gb + 4 * PRE + s];
    for (int c = 0; c < 4; ++c) {
      sicx[lane][c] = geom[gb + (5 + c) * PRE + s];
      sicy[lane][c] = geom[gb + (9 + c) * PRE + s];
    }
  } else {
    int l2 = lane - 16;
    int s = j0 + l2;
    sjx[l2] = geom[gb + 0 * PRE + s];
    sjy[l2] = geom[gb + 1 * PRE + s];
    sjq[l2] = geom[gb + 2 * PRE + s];
    sjr[l2] = geom[gb + 3 * PRE + s];
    sja[l2] = geom[gb + 4 * PRE + s];
    for (int c = 0; c < 4; ++c) {
      sjcx[l2][c] = geom[gb + (5 + c) * PRE + s];
      sjcy[l2][c] = geom[gb + (9 + c) * PRE + s];
    }
  }
  __syncthreads();

  int lm = lane & 15;
  int kb = (lane >> 4) * 2;     // A: VGPR0=K{0|2}, VGPR1=K{1|3} per half-wave
  float fa[4] = { six[lm], siy[lm], siq[lm], 1.0f };
  float fb[4] = { -2.f * sjx[lm], -2.f * sjy[lm], 1.0f, sjq[lm] };
  v2f A;  A.x  = fa[kb]; A.y  = fa[kb + 1];
  v2f Bf; Bf.x = fb[kb]; Bf.y = fb[kb + 1];
  v8f C = {};
#if __has_builtin(__builtin_amdgcn_wmma_f32_16x16x4_f32)
  v8f D = __builtin_amdgcn_wmma_f32_16x16x4_f32(false, A, false, Bf, (short)0, C, false, false);
#else
  v8f D = C;  // (fallback: dist2=1e30 fails the prefilter -> no pair clipped)
  #pragma unroll
  for (int q = 0; q < 8; ++q) D[q] = 1e30f;
#endif
  float dd[8];
#pragma unroll
  for (int q = 0; q < 8; ++q) dd[q] = D[q];

  for (int r = 0; r < 8; ++r) {
    int mi = (lane < 16) ? r : (8 + r);  // C/D layout: lanes0-15 M=r, lanes16-31 M=8+r
    int i = i0 + mi, j = j0 + lm;
    bool flag = false;
    if (j > i) {
      float dist2 = dd[r];
      float sr = sir[mi] + sjr[lm];
      if (dist2 <= sr * sr) {
        float inter = interArea(&sicx[mi][0], &sicy[mi][0], &sjcx[lm][0], &sjcy[lm][0]);
        float uni = sia[mi] + sja[lm] - inter;
        flag = (inter / fmaxf(uni, 1e-8f)) > IOU_TH;
      }
    }
    unsigned long long bal = __ballot((int)flag);
    if (lane == 0) {
      unsigned bits = (unsigned)(bal & 0xFFFFull);
      if (bits) atomicOr(&mask[(size_t)(p * PRE + i0 + r) * 16 + (j0 >> 5)], bits << (j0 & 31));
    }
    if (lane == 16) {
      unsigned bits = (unsigned)((bal >> 16) & 0xFFFFull);
      if (bits) atomicOr(&mask[(size_t)(p * PRE + i0 + 8 + r) * 16 + (j0 >> 5)], bits << (j0 & 31));
    }
  }
}

// ---------- sequential greedy suppression, 1 wave per problem ----------
__global__ void k_supp(const unsigned* __restrict__ mask, int* __restrict__ keepPos,
                       int* __restrict__ nKeep) {
  int p = blockIdx.x;
  int l = threadIdx.x;
  unsigned supp = 0u;
  for (int i = 0; i < PRE; ++i) {
    unsigned sw = (unsigned)__shfl((int)supp, i >> 5, 32);
    if (!((sw >> (i & 31)) & 1u)) {
      unsigned row = (l < 16) ? mask[(size_t)(p * PRE + i) * 16 + l] : 0u;
      supp |= row;   // mask only contains j>i bits by construction
    }
  }
  unsigned keep = (l < 16) ? ~supp : 0u;
  int cnt = __popc(keep);
  int pre = 0, tot = 0;
  for (int s = 0; s < 16; ++s) {
    int c = __shfl(cnt, s, 32);
    tot += c;
    if (s < l) pre += c;
  }
  unsigned m = keep;
  int r = pre;
  while (m) {
    int bp = __ffs((int)m) - 1;
    if (r < POST) keepPos[p * 128 + r] = l * 32 + bp;
    m &= m - 1;
    ++r;
  }
  if (l == 0) nKeep[p] = tot;
}

// ---------- final output: boxes+score, labels, valid ----------
__global__ void k_out(const float* __restrict__ box7, const int* __restrict__ topI,
                      const int* __restrict__ keepPos, const int* __restrict__ nKeep,
                      const float* __restrict__ tot, const float* __restrict__ res,
                      int N, float* __restrict__ out) {
  int t = blockIdx.x * blockDim.x + threadIdx.x;
  if (t >= NPROB * POST) return;
  int p = t / POST, k = t % POST;
  int v = p >> 2, b = p & 3;
  int nk = nKeep[p]; if (nk > POST) nk = POST;
  float o[8] = {0, 0, 0, 0, 0, 0, 0, 0};
  float val = 0.f;
  if (k < nk) {
    int s = keepPos[p * 128 + k];
    const float* bx = box7 + (size_t)(p * PRE + s) * 7;
    float x = bx[0], y = bx[1], z = bx[2];
    bool inr = (x >= 0.f) && (x <= 70.4f) && (y >= -40.f) && (y <= 40.f) &&
               (z >= -2.2f) && (z <= 0.8f);
    if (inr) {
      for (int c = 0; c < 7; ++c) o[c] = bx[c];
      int row = ((v == 3) ? 1 : 0) * 4 + b;
      int idx = topI[row * PRE + s];
      o[7] = (v < 3) ? tot[(size_t)b * N + idx] : res[(size_t)b * N + idx];
      val = 1.f;
    }
  }
  float* om = out + (size_t)(p * POST + k) * 8;
  for (int c = 0; c < 8; ++c) om[c] = o[c];
  out[NPROB * POST * 8 + p * POST + k] = 0.f;   // labels
  out[NPROB * POST * 9 + p * POST + k] = val;   // valid
}

extern "C" void kernel_launch(void* const* d_in, const int* in_sizes, int n_in,
                              void* d_out, int out_size, void* d_ws, size_t ws_size,
                              hipStream_t stream) {
  const float* logits = (const float*)d_in[0];
  const float* breg   = (const float*)d_in[1];
  const float* anch   = (const float*)d_in[2];
  const float* res    = (const float*)d_in[3];
  int BN = in_sizes[0];
  int N  = BN / BB;

  float* ws = (float*)d_ws;
  size_t oTot = 0;
  size_t oH1  = oTot + (size_t)BN;
  size_t oH2  = oH1 + 2048;
  size_t oCnt = oH2 + 32768;
  size_t oT1  = oCnt + 8;
  size_t oCA  = oT1 + 8;
  size_t oT2  = oCA + 8;
  size_t oNK  = oT2 + 8;
  size_t oCK  = oNK + 16;
  size_t oCI  = oCK + (size_t)8 * CAND;
  size_t oTop = oCI + (size_t)8 * CAND;
  size_t oBox = oTop + (size_t)8 * PRE;
  size_t oGeo = oBox + (size_t)NPROB * PRE * 7;
  size_t oMsk = oGeo + (size_t)NPROB * PRE * 13;
  size_t oKp  = oMsk + (size_t)NPROB * PRE * 16;

  float*    tot   = ws + oTot;
  unsigned* hist1 = (unsigned*)(ws + oH1);
  unsigned* hist2 = (unsigned*)(ws + oH2);
  int*      cnt   = (int*)(ws + oCnt);
  int*      t1    = (int*)(ws + oT1);
  int*      cA    = (int*)(ws + oCA);
  int*      t2    = (int*)(ws + oT2);
  int*      nK    = (int*)(ws + oNK);
  unsigned* ckey  = (unsigned*)(ws + oCK);
  int*      cidx  = (int*)(ws + oCI);
  int*      topI  = (int*)(ws + oTop);
  float*    box7  = ws + oBox;
  float*    geom  = ws + oGeo;
  unsigned* mask  = (unsigned*)(ws + oMsk);
  int*      keepP = (int*)(ws + oKp);

  // zero: hist1..nKeep block (contiguous) and the IOU bitmask
  k_zero<<<64, 256, 0, stream>>>(hist1, 2048 + 32768 + 8 + 8 + 8 + 8 + 16);
  k_zero<<<256, 256, 0, stream>>>(mask, NPROB * PRE * 16);

  k_scores<<<1024, 256, 0, stream>>>(logits, tot, BN);

  dim3 gh(128, 8);
  k_hist1<<<gh, 256, 0, stream>>>(tot, res, N, hist1);
  k_thresh1<<<8, 1, 0, stream>>>(hist1, t1, cA);
  k_hist2<<<gh, 256, 0, stream>>>(tot, res, N, t1, hist2);
  k_thresh2<<<8, 1, 0, stream>>>(hist2, cA, t2);
  k_compact<<<gh, 256, 0, stream>>>(tot, res, N, t1, t2, cnt, ckey, cidx);
  k_sort<<<8, 1024, 0, stream>>>(ckey, cidx, cnt, topI);

  k_decode<<<(NPROB * PRE + 255) / 256, 256, 0, stream>>>(topI, breg, anch, N, box7, geom);
  k_iou<<<NPROB * 32 * 32, 32, 0, stream>>>(geom, mask);
  k_supp<<<NPROB, 32, 0, stream>>>(mask, keepP, nK);
  k_out<<<(NPROB * POST + 255) / 256, 256, 0, stream>>>(box7, topI, keepP, nK, tot, res, N,
                                                        (float*)d_out);
}